// StochasticMultiheadAttention_79809082294280
// MI455X (gfx1250) — compile-verified
//
#include <hip/hip_runtime.h>

// ---- problem constants (from reference) ------------------------------------
#define S_LEN 2048
#define BSZ   4
#define EMB   512
#define NHEAD 8
#define DH    64
#define BH    (BSZ * NHEAD)   // 32
#define E3    (3 * EMB)       // 1536
#define QKV_ROW (BSZ * E3)    // row stride (in floats) of qkv[s][b][3E] along s

typedef __attribute__((ext_vector_type(2))) float v2f;
typedef __attribute__((ext_vector_type(8))) float v8f;

// One f32 WMMA step: C(16x16) += A(16x4) * B(4x16), wave32, even-VGPR operands.
// A layout: lanes 0-15 -> M=lane, VGPR0/1 = K=0/1 ; lanes 16-31 -> K=2/3.
// B layout (mirror): lanes 0-15 -> N=lane, VGPR0/1 = K=0/1 ; lanes 16-31 -> K=2/3.
// C layout: lanes 0-15 -> N=lane, VGPR v = M=v ; lanes 16-31 -> M=v+8.
__device__ __forceinline__ v8f wmma_f32_step(v2f a, v2f b, v8f c) {
    return __builtin_amdgcn_wmma_f32_16x16x4_f32(
        /*neg_a=*/false, a, /*neg_b=*/false, b,
        /*c_mod=*/(short)0, c, /*reuse_a=*/false, /*reuse_b=*/false);
}

// ---------------------------------------------------------------------------
// Generic GEMM with bias: Out[M][ldo] = A[M][K] * W[N][K]^T + bias[N]
// One wave per 16x16 output tile; blockDim = 256 (8 waves).
// ---------------------------------------------------------------------------
__global__ void gemm_bias_wmma(const float* __restrict__ A, int lda,
                               const float* __restrict__ W, int ldw,
                               const float* __restrict__ bias,
                               float* __restrict__ Out, int ldo,
                               int ntiles_n, int K)
{
    const int wave = (blockIdx.x * blockDim.x + threadIdx.x) >> 5;
    const int lane = threadIdx.x & 31;
    const int mt = wave / ntiles_n;
    const int nt = wave - mt * ntiles_n;
    const int mn = lane & 15;            // row-in-tile for A / col-in-tile for B,C
    const int kk = (lane >> 4) << 1;     // 0 or 2

    const float* arow = A + (size_t)(mt * 16 + mn) * lda;
    const float* wrow = W + (size_t)(nt * 16 + mn) * ldw;

    v8f c = {0.f, 0.f, 0.f, 0.f, 0.f, 0.f, 0.f, 0.f};
    for (int k0 = 0; k0 < K; k0 += 4) {
        v2f a = *(const v2f*)(arow + k0 + kk);
        v2f b = *(const v2f*)(wrow + k0 + kk);
        c = wmma_f32_step(a, b, c);
    }

    const int ncol  = nt * 16 + mn;
    const float bv  = bias ? bias[ncol] : 0.f;
    const int mbase = mt * 16 + ((lane >> 4) << 3);
#pragma unroll
    for (int i = 0; i < 8; ++i)
        Out[(size_t)(mbase + i) * ldo + ncol] = c[i] + bv;
}

// ---------------------------------------------------------------------------
// Fused attention for one (head, 16-row query tile):
//   scores (WMMA f32) -> LDS (16x2048, 128 KB) -> softmax in LDS ->
//   single write of normalized weights to HBM -> P@V (WMMA f32) from LDS.
// blockDim = 256 (8 waves). Needs ~145 KB LDS (CDNA5: 320 KB/WGP).
// ---------------------------------------------------------------------------
__global__ void __launch_bounds__(256) attn_fused_wmma(
        const float* __restrict__ qkv,   // [S][B][3E]
        float* __restrict__ wout,        // attn_weights [BH][S][S]
        float* __restrict__ opre)        // [S][B][E]
{
    __shared__ float sS[16 * S_LEN];     // score / P strip, 128 KB
    __shared__ float sQ[16 * DH];        // Q tile, 4 KB
    __shared__ float sRed[16 * 16];
    __shared__ float sMax[16];
    __shared__ float sSum[16];
    __shared__ float sO[8 * 16 * 16];    // per-wave partial O tiles, 8 KB

    const int bh = blockIdx.x >> 7;      // / (S/16 = 128)
    const int it = blockIdx.x & 127;
    const int i0 = it * 16;
    const int b  = bh >> 3;
    const int h  = bh & 7;

    const float* qbase = qkv + b * E3 + h * DH;          // + s*QKV_ROW + d
    const float* kbase = qbase + EMB;
    const float* vbase = qbase + 2 * EMB;

    const int tid  = threadIdx.x;
    const int wv   = tid >> 5;
    const int lane = tid & 31;
    const int mn   = lane & 15;
    const int kk   = (lane >> 4) << 1;

    // ---- load Q tile (16 x 64) into LDS ----
    for (int idx = tid; idx < 16 * DH; idx += 256) {
        int r = idx >> 6, d = idx & 63;
        sQ[idx] = qbase[(size_t)(i0 + r) * QKV_ROW + d];
    }
    __syncthreads();

    // ---- phase 1: S = Q * K^T, 16 WMMAs per 16x16 tile, tiles split over waves
    for (int jt = wv; jt < S_LEN / 16; jt += 8) {
        const int j0 = jt * 16;
        const float* krow = kbase + (size_t)(j0 + mn) * QKV_ROW;
        if (jt + 8 < S_LEN / 16)
            __builtin_prefetch(kbase + (size_t)(j0 + 128 + mn) * QKV_ROW, 0, 1);
        v8f c = {0.f, 0.f, 0.f, 0.f, 0.f, 0.f, 0.f, 0.f};
#pragma unroll
        for (int k0 = 0; k0 < DH; k0 += 4) {
            v2f a  = *(const v2f*)(sQ + mn * DH + k0 + kk);
            v2f bb = *(const v2f*)(krow + k0 + kk);
            c = wmma_f32_step(a, bb, c);
        }
        const int row0 = (lane >> 4) << 3;
#pragma unroll
        for (int i = 0; i < 8; ++i)
            sS[(row0 + i) * S_LEN + j0 + mn] = c[i];
    }
    __syncthreads();

    // ---- phase 2: softmax over each of the 16 rows (2048 wide), in LDS ----
    // thread (r, seg): r = tid>>4 handles row r, columns seg + 16*q (coalesced).
    const int r = tid >> 4, seg = tid & 15;
    float mx = -3.402823466e38f;
    for (int q = 0; q < 128; ++q)
        mx = fmaxf(mx, sS[r * S_LEN + seg + (q << 4)]);
    sRed[r * 16 + seg] = mx;
    __syncthreads();
    if (tid < 16) {
        float m2 = sRed[tid * 16];
        for (int i = 1; i < 16; ++i) m2 = fmaxf(m2, sRed[tid * 16 + i]);
        sMax[tid] = m2;
    }
    __syncthreads();
    const float rmax = sMax[r];
    float sum = 0.f;
    for (int q = 0; q < 128; ++q) {
        const int idx = r * S_LEN + seg + (q << 4);
        const float e = __expf(sS[idx] - rmax);
        sS[idx] = e;
        sum += e;
    }
    sRed[r * 16 + seg] = sum;
    __syncthreads();
    if (tid < 16) {
        float s2 = 0.f;
        for (int i = 0; i < 16; ++i) s2 += sRed[tid * 16 + i];
        sSum[tid] = s2;
    }
    __syncthreads();
    const float inv = 1.0f / sSum[r];
    float* wrow = wout + ((size_t)bh * S_LEN + i0 + r) * S_LEN;
    for (int q = 0; q < 128; ++q) {
        const int cix = seg + (q << 4);
        const float p = sS[r * S_LEN + cix] * inv;
        sS[r * S_LEN + cix] = p;
        wrow[cix] = p;            // the one-and-only HBM pass over attn_weights
    }
    __syncthreads();

    // ---- phase 3: O = P(16x2048) @ V(2048x64). 8 waves: n-tile = wv&3,
    //      k-half = wv>>2; combine the two k-halves through LDS. ----
    {
        const int ntile = wv & 3, khalf = wv >> 2;
        const int n0 = ntile * 16;
        const int kbeg = khalf * (S_LEN / 2), kend = kbeg + (S_LEN / 2);
        v8f c = {0.f, 0.f, 0.f, 0.f, 0.f, 0.f, 0.f, 0.f};
        for (int k0 = kbeg; k0 < kend; k0 += 4) {
            v2f a = *(const v2f*)(sS + mn * S_LEN + k0 + kk);
            v2f bb;
            bb.x = vbase[(size_t)(k0 + kk)     * QKV_ROW + n0 + mn];
            bb.y = vbase[(size_t)(k0 + kk + 1) * QKV_ROW + n0 + mn];
            c = wmma_f32_step(a, bb, c);
        }
        const int row0 = (lane >> 4) << 3;
#pragma unroll
        for (int i = 0; i < 8; ++i)
            sO[wv * 256 + (row0 + i) * 16 + mn] = c[i];
    }
    __syncthreads();

    // combine k-halves and scatter to opre[s][b][E] layout
    for (int idx = tid; idx < 16 * 64; idx += 256) {
        const int rr = idx >> 6;          // row in tile
        const int cc = idx & 63;          // col in Dh
        const int nt2 = cc >> 4, nn = cc & 15;
        const float val = sO[nt2 * 256 + rr * 16 + nn]
                        + sO[(nt2 + 4) * 256 + rr * 16 + nn];
        opre[((size_t)(i0 + rr) * BSZ + b) * EMB + h * DH + cc] = val;
    }
}

// ---------------------------------------------------------------------------
extern "C" void kernel_launch(void* const* d_in, const int* in_sizes, int n_in,
                              void* d_out, int out_size, void* d_ws, size_t ws_size,
                              hipStream_t stream) {
    const float* query = (const float*)d_in[0];
    // d_in[1] (key) and d_in[2] (value) are unused: reference's
    // _qkv_same_embed_dim branch projects q, k, v all from `query`.
    const float* in_w  = (const float*)d_in[3];   // [3E, E]
    const float* in_b  = (const float*)d_in[4];   // [3E]
    const float* out_w = (const float*)d_in[5];   // [E, E]
    const float* out_b = (const float*)d_in[6];   // [E]

    float* out       = (float*)d_out;
    float* attn_out  = out;                                   // [S,B,E]
    float* attn_w    = out + (size_t)S_LEN * BSZ * EMB;       // [B,H,S,S]

    float* qkv  = (float*)d_ws;                               // [S*B][3E]  (48 MB)
    float* opre = qkv + (size_t)S_LEN * BSZ * E3;             // [S*B][E]   (16 MB)

    // 1) QKV projection: [8192,512] x [1536,512]^T -> [8192,1536]
    {
        const int mt = (S_LEN * BSZ) / 16;   // 512
        const int nt = E3 / 16;              // 96
        gemm_bias_wmma<<<(mt * nt) / 8, 256, 0, stream>>>(
            query, EMB, in_w, EMB, in_b, qkv, E3, nt, EMB);
    }

    // 2) fused attention: scores + softmax + weights write + P@V
    attn_fused_wmma<<<BH * (S_LEN / 16), 256, 0, stream>>>(qkv, attn_w, opre);

    // 3) output projection: [8192,512] x [512,512]^T -> [8192,512]
    {
        const int mt = (S_LEN * BSZ) / 16;   // 512
        const int nt = EMB / 16;             // 32
        gemm_bias_wmma<<<(mt * nt) / 8, 256, 0, stream>>>(
            opre, EMB, out_w, EMB, out_b, attn_out, EMB, nt, EMB);
    }
}